// CustomGAT_7799660609771
// MI455X (gfx1250) — compile-verified
//
#include <hip/hip_runtime.h>
#include <math.h>

#define N_NODES 50000
#define IN_CH   256
#define HEADS   8
#define OUT_CH  32
#define OUT_FEAT 256              // HEADS*OUT_CH
#define N_EDGES 800000
#define ETOT    (N_EDGES + N_NODES)
#define COLTILES_PER_WAVE 8       // 128 output columns per wave

typedef float v2f __attribute__((ext_vector_type(2)));
typedef float v8f __attribute__((ext_vector_type(8)));

// ---- ordered-uint encoding of float so unsigned atomicMax == float max ----
__device__ __forceinline__ unsigned f2ord(float f) {
  unsigned u = __float_as_uint(f);
  return (u & 0x80000000u) ? ~u : (u | 0x80000000u);
}
__device__ __forceinline__ float ord2f(unsigned u) {
  return (u & 0x80000000u) ? __uint_as_float(u & 0x7FFFFFFFu)
                           : __uint_as_float(~u);
}
__device__ __forceinline__ void atomicAddF32(float* p, float v) {
  unsafeAtomicAdd(p, v);   // global_atomic_add_f32 on CDNA5
}
__device__ __forceinline__ float lrelu02(float a) {
  return a > 0.f ? a : 0.2f * a;
}

// ---------------- K0: init out, alpha_sum, per-head max ----------------
__global__ void gat_init(float* __restrict__ out, float* __restrict__ asum,
                         unsigned* __restrict__ maxbits) {
  size_t i = (size_t)blockIdx.x * blockDim.x + threadIdx.x;
  size_t stride = (size_t)gridDim.x * blockDim.x;
  for (size_t p = i; p < (size_t)N_NODES * OUT_FEAT; p += stride) out[p] = 0.f;
  for (size_t p = i; p < (size_t)N_NODES * HEADS; p += stride) asum[p] = 0.f;
  if (i < HEADS) maxbits[i] = f2ord(-1e30f);
}

// ---------------- K1: xp = x @ W^T with V_WMMA_F32_16X16X4_F32 ----------------
// One wave per 16-row x 128-col strip: each A fragment load feeds 8 WMMAs,
// cutting x (A-matrix) L2 traffic 8x vs one-tile-per-wave.
__global__ __launch_bounds__(32) void gat_gemm_wmma(const float* __restrict__ x,
                                                    const float* __restrict__ W,
                                                    float* __restrict__ xp) {
  const int lane  = threadIdx.x;                       // one wave32 per block
  const int colg0 = blockIdx.x * (16 * COLTILES_PER_WAVE);
  const int row0  = blockIdx.y * 16;
  const int m     = lane & 15;                         // A row / B col within tile
  const int khalf = (lane >> 4) << 1;                  // 0 (lanes 0-15) or 2

  const float2* xr = (const float2*)x + (size_t)(row0 + m) * (IN_CH / 2);
  const float2* wb = (const float2*)W + (size_t)(colg0 + m) * (IN_CH / 2);

  v8f acc[COLTILES_PER_WAVE];
  #pragma unroll
  for (int j = 0; j < COLTILES_PER_WAVE; ++j) acc[j] = (v8f){};

  for (int k0 = 0; k0 < IN_CH; k0 += 4) {
    const int kp = (k0 + khalf) >> 1;
    const float2 av = xr[kp];                          // A[m][khalf..khalf+1]
    v2f a; a.x = av.x; a.y = av.y;
    #pragma unroll
    for (int j = 0; j < COLTILES_PER_WAVE; ++j) {
      // B[k][n] = W[n][k]; lane's column in tile j is colg0 + j*16 + m
      const float2 bv = wb[(size_t)j * 16 * (IN_CH / 2) + kp];
      v2f b; b.x = bv.x; b.y = bv.y;
      acc[j] = __builtin_amdgcn_wmma_f32_16x16x4_f32(
          /*neg_a=*/false, a, /*neg_b=*/false, b,
          /*c_mod=*/(short)0, acc[j], /*reuse_a=*/false, /*reuse_b=*/false);
    }
  }
  // C/D layout: VGPR r -> M=r (lanes 0-15) / M=r+8 (lanes 16-31), N=lane&15
  const int rowadd = (lane >> 4) << 3;
  #pragma unroll
  for (int j = 0; j < COLTILES_PER_WAVE; ++j)
    #pragma unroll
    for (int r = 0; r < 8; ++r)
      xp[(size_t)(row0 + r + rowadd) * OUT_FEAT + colg0 + j * 16 + m] = acc[j][r];
}

// ---------------- K2: per-node per-head attention dots ----------------
__global__ void gat_node_alpha(const float* __restrict__ xp,
                               const float* __restrict__ attn_l,
                               const float* __restrict__ attn_r,
                               float* __restrict__ al, float* __restrict__ ar) {
  int idx = blockIdx.x * blockDim.x + threadIdx.x;   // node*HEADS + h
  if (idx >= N_NODES * HEADS) return;
  const int h = idx & (HEADS - 1);
  const float* v  = xp + (size_t)idx * OUT_CH;       // node*256 + h*32
  const float* wl = attn_l + h * OUT_CH;
  const float* wr = attn_r + h * OUT_CH;
  float sl = 0.f, sr = 0.f;
  #pragma unroll
  for (int c = 0; c < OUT_CH; ++c) { float t = v[c]; sl += t * wl[c]; sr += t * wr[c]; }
  al[idx] = sl; ar[idx] = sr;
}

__device__ __forceinline__ void edge_endpoints(const long long* __restrict__ ei,
                                               int e, int& src, int& dst) {
  if (e < N_EDGES) { src = (int)ei[e]; dst = (int)ei[N_EDGES + e]; }
  else             { src = dst = e - N_EDGES; }       // self loops
}

// ---------------- K3: global per-head max over edge logits ----------------
__global__ __launch_bounds__(256) void gat_edge_max(const long long* __restrict__ ei,
                                                    const float* __restrict__ al,
                                                    const float* __restrict__ ar,
                                                    unsigned* __restrict__ maxbits) {
  __shared__ float smax[8][HEADS];                    // [wave][head]
  const int e = blockIdx.x * blockDim.x + threadIdx.x;
  float mv[HEADS];
  #pragma unroll
  for (int h = 0; h < HEADS; ++h) mv[h] = -1e30f;
  if (e < ETOT) {
    int src, dst; edge_endpoints(ei, e, src, dst);
    #pragma unroll
    for (int h = 0; h < HEADS; ++h)
      mv[h] = lrelu02(al[src * HEADS + h] + ar[dst * HEADS + h]);
  }
  #pragma unroll
  for (int off = 16; off > 0; off >>= 1)
    #pragma unroll
    for (int h = 0; h < HEADS; ++h)
      mv[h] = fmaxf(mv[h], __shfl_xor(mv[h], off, 32));
  const int wave = threadIdx.x >> 5, lane = threadIdx.x & 31;
  if (lane < HEADS) smax[wave][lane] = mv[lane];
  __syncthreads();
  if (threadIdx.x < HEADS) {
    float m = smax[0][threadIdx.x];
    #pragma unroll
    for (int w = 1; w < 8; ++w) m = fmaxf(m, smax[w][threadIdx.x]);
    atomicMax(&maxbits[threadIdx.x], f2ord(m));
  }
}

// ---------------- K4: exp(logit - max) segment-sum over dst ----------------
__global__ __launch_bounds__(256) void gat_edge_expsum(const long long* __restrict__ ei,
                                                       const float* __restrict__ al,
                                                       const float* __restrict__ ar,
                                                       const unsigned* __restrict__ maxbits,
                                                       float* __restrict__ asum) {
  const int e = blockIdx.x * blockDim.x + threadIdx.x;
  if (e >= ETOT) return;
  int src, dst; edge_endpoints(ei, e, src, dst);
  #pragma unroll
  for (int h = 0; h < HEADS; ++h) {
    float a = lrelu02(al[src * HEADS + h] + ar[dst * HEADS + h]);
    float ex = __expf(a - ord2f(maxbits[h]));
    atomicAddF32(&asum[dst * HEADS + h], ex);
  }
}

// ---------------- K5: normalize + weighted scatter-add ----------------
// One 256-thread block per edge; thread t handles feature column t (h = t/32).
__global__ __launch_bounds__(256) void gat_aggregate(const long long* __restrict__ ei,
                                                     const float* __restrict__ al,
                                                     const float* __restrict__ ar,
                                                     const unsigned* __restrict__ maxbits,
                                                     const float* __restrict__ asum,
                                                     const float* __restrict__ xp,
                                                     float* __restrict__ out) {
  const int e = blockIdx.x;
  const int t = threadIdx.x;
  const int h = t >> 5;
  int src, dst; edge_endpoints(ei, e, src, dst);
  float a  = lrelu02(al[src * HEADS + h] + ar[dst * HEADS + h]);
  float ex = __expf(a - ord2f(maxbits[h]));
  float wgt = ex / fmaxf(asum[dst * HEADS + h], 1e-6f);
  float val = wgt * xp[(size_t)src * OUT_FEAT + t];
  atomicAddF32(&out[(size_t)dst * OUT_FEAT + t], val);
}

extern "C" void kernel_launch(void* const* d_in, const int* in_sizes, int n_in,
                              void* d_out, int out_size, void* d_ws, size_t ws_size,
                              hipStream_t stream) {
  const float*     x      = (const float*)d_in[0];
  const long long* ei     = (const long long*)d_in[1];   // int64 edge_index [2,E]
  const float*     W      = (const float*)d_in[2];
  const float*     attn_l = (const float*)d_in[3];
  const float*     attn_r = (const float*)d_in[4];
  float*           out    = (float*)d_out;

  // workspace carve-up (~56 MB)
  char* ws = (char*)d_ws;
  float* xp   = (float*)ws; ws += sizeof(float) * (size_t)N_NODES * OUT_FEAT;
  float* al   = (float*)ws; ws += sizeof(float) * (size_t)N_NODES * HEADS;
  float* ar   = (float*)ws; ws += sizeof(float) * (size_t)N_NODES * HEADS;
  float* asum = (float*)ws; ws += sizeof(float) * (size_t)N_NODES * HEADS;
  unsigned* maxbits = (unsigned*)ws;

  gat_init<<<2048, 256, 0, stream>>>(out, asum, maxbits);
  gat_gemm_wmma<<<dim3(OUT_FEAT / (16 * COLTILES_PER_WAVE), N_NODES / 16),
                  32, 0, stream>>>(x, W, xp);
  gat_node_alpha<<<(N_NODES * HEADS + 255) / 256, 256, 0, stream>>>(xp, attn_l, attn_r, al, ar);
  gat_edge_max<<<(ETOT + 255) / 256, 256, 0, stream>>>(ei, al, ar, maxbits);
  gat_edge_expsum<<<(ETOT + 255) / 256, 256, 0, stream>>>(ei, al, ar, maxbits, asum);
  gat_aggregate<<<ETOT, 256, 0, stream>>>(ei, al, ar, maxbits, asum, xp, out);
}